// Switcher_33328946217459
// MI455X (gfx1250) — compile-verified
//
#include <hip/hip_runtime.h>
#include <hip/hip_bf16.h>

// ---- problem constants (match reference) ----
#define T_TOK   131072
#define DIM     1024
#define NEXP    64
#define CAP     2560          // int(1.25 * 131072 / 64)
#define ALPHA_C 0.01f

typedef float v2f __attribute__((ext_vector_type(2)));
typedef float v8f __attribute__((ext_vector_type(8)));

// =====================================================================
// Kernel A: fused  logits = x@W + b  (WMMA f32 16x16x4)  -> softmax ->
// top-1 expert index, per-block expert histogram, per-block Pi partials.
// Block = 256 threads (8 waves), each wave owns a 16-row x 64-col tile.
// W tile is staged K-pair-interleaved so each B fragment is ONE aligned
// ds_load_b64 (no cross-pitch gather + v_mov repack).
// =====================================================================
__global__ __launch_bounds__(256) void moe_router_gemm(
    const float* __restrict__ x, const float* __restrict__ W,
    const float* __restrict__ b,
    int* __restrict__ eIdx, int* __restrict__ chunkCounts,
    float* __restrict__ piPart)
{
    __shared__ __align__(16) float xs[128][68];     // pitch 68: 16B rows, conflict-free A reads
    // wpair[g][h][col'*2+q] = W[kk + 4g + 2h + q][col'] ; row stride 160
    // (160 % 64 == 32 -> the two half-wave groups use disjoint bank halves)
    __shared__ __align__(16) float wpair[16][2][160];
    __shared__ float piW[8][64];
    __shared__ int   hist[64];

    const int tid  = threadIdx.x;
    const int wave = tid >> 5, lane = tid & 31;
    const int col  = lane & 15, half = lane >> 4;
    const int koff = half * 2;            // K sub-offset per ISA A 16x4 f32 layout
    const int row0 = blockIdx.x * 128;

    if (tid < 64) hist[tid] = 0;

    v8f acc[4] = {};                      // 4 col-tiles of 16 -> N=64

    const float4* x4 = reinterpret_cast<const float4*>(x);
    const float4* W4 = reinterpret_cast<const float4*>(W);

    for (int kk = 0; kk < DIM; kk += 64) {
        __syncthreads();
        // stage x tile: 128 rows x 64 cols = 2048 float4, coalesced
        #pragma unroll
        for (int i = 0; i < 8; ++i) {
            int f = tid + i * 256;
            int r = f >> 4, c4 = f & 15;
            float4 v = x4[(size_t)(row0 + r) * (DIM / 4) + (kk >> 2) + c4];
            *reinterpret_cast<float4*>(&xs[r][c4 * 4]) = v;
        }
        // stage W tile (64 k-rows x 64 cols), scattered into K-pair layout
        #pragma unroll
        for (int i = 0; i < 4; ++i) {
            int f = tid + i * 256;
            int k = f >> 4, c4 = f & 15;
            float4 v = W4[(size_t)(kk + k) * (NEXP / 4) + c4];
            int g = k >> 2, h = (k >> 1) & 1, q = k & 1;
            wpair[g][h][(c4 * 4 + 0) * 2 + q] = v.x;
            wpair[g][h][(c4 * 4 + 1) * 2 + q] = v.y;
            wpair[g][h][(c4 * 4 + 2) * 2 + q] = v.z;
            wpair[g][h][(c4 * 4 + 3) * 2 + q] = v.w;
        }
        __syncthreads();

        const int rl = wave * 16 + col;   // A row for this lane
        #pragma unroll
        for (int g = 0; g < 16; ++g) {    // k2 = 4g
            v2f a;
            a.x = xs[rl][g * 4 + koff];
            a.y = xs[rl][g * 4 + koff + 1];
            #pragma unroll
            for (int c = 0; c < 4; ++c) {
                v2f bb = *reinterpret_cast<const v2f*>(
                    &wpair[g][half][(c * 16 + col) * 2]);
                acc[c] = __builtin_amdgcn_wmma_f32_16x16x4_f32(
                    false, a, false, bb, (short)0, acc[c], false, false);
            }
        }
    }

    // bias
    float bias[4];
    #pragma unroll
    for (int c = 0; c < 4; ++c) bias[c] = b[c * 16 + col];

    // per-row softmax / argmax. C/D layout: component v of acc[c] holds
    // row (v + 8*half), col (c*16 + lane%16).
    float piAcc[4] = {0.f, 0.f, 0.f, 0.f};
    #pragma unroll
    for (int v = 0; v < 8; ++v) {
        float vals[4];
        #pragma unroll
        for (int c = 0; c < 4; ++c) vals[c] = acc[c][v] + bias[c];

        float m = fmaxf(fmaxf(vals[0], vals[1]), fmaxf(vals[2], vals[3]));
        #pragma unroll
        for (int off = 1; off < 16; off <<= 1) m = fmaxf(m, __shfl_xor(m, off));

        float ex[4], s = 0.f;
        #pragma unroll
        for (int c = 0; c < 4; ++c) { ex[c] = __expf(vals[c] - m); s += ex[c]; }
        #pragma unroll
        for (int off = 1; off < 16; off <<= 1) s += __shfl_xor(s, off);
        float inv = 1.f / s;

        // argmax with lowest-index tie-break (matches lax.top_k stability)
        float bv = vals[0]; int bi = col;
        #pragma unroll
        for (int c = 1; c < 4; ++c) {
            int idx = c * 16 + col;
            if (vals[c] > bv) { bv = vals[c]; bi = idx; }
        }
        #pragma unroll
        for (int off = 1; off < 16; off <<= 1) {
            float ov = __shfl_xor(bv, off);
            int   oi = __shfl_xor(bi, off);
            if (ov > bv || (ov == bv && oi < bi)) { bv = ov; bi = oi; }
        }

        #pragma unroll
        for (int c = 0; c < 4; ++c) piAcc[c] += ex[c] * inv;

        if (col == 0) {
            int rowg = row0 + wave * 16 + v + 8 * half;
            eIdx[rowg] = bi;
            atomicAdd(&hist[bi], 1);      // integer counts: deterministic
        }
    }

    // combine the two half-wave row groups (same columns) deterministically
    #pragma unroll
    for (int c = 0; c < 4; ++c) {
        float other = __shfl_xor(piAcc[c], 16);
        if (half == 0) piW[wave][c * 16 + col] = piAcc[c] + other;
    }
    __syncthreads();
    if (tid < 64) {
        float s = 0.f;
        #pragma unroll
        for (int w = 0; w < 8; ++w) s += piW[w][tid];   // fixed order
        piPart[(size_t)blockIdx.x * 64 + tid]      = s;
        chunkCounts[(size_t)blockIdx.x * 64 + tid] = hist[tid];
    }
}

// =====================================================================
// Kernel B: per-expert exclusive scan over 1024 chunks, Pi reduction,
// aux loss, capacity clamp. Single block of 64 threads, fixed order.
// =====================================================================
__global__ __launch_bounds__(64) void moe_scan(
    const int* __restrict__ chunkCounts, const float* __restrict__ piPart,
    int* __restrict__ chunkBase, int* __restrict__ filled,
    float* __restrict__ lossOut)
{
    __shared__ float lt[64];
    int e = threadIdx.x;

    float pi = 0.f;
    for (int blk = 0; blk < 1024; ++blk) pi += piPart[blk * 64 + e];

    int base = 0;
    for (int blk = 0; blk < 1024; ++blk) {
        chunkBase[blk * 64 + e] = base;
        base += chunkCounts[blk * 64 + e];
    }
    filled[e] = base < CAP ? base : CAP;

    float fi = (float)base / (float)T_TOK;
    float Pi = pi / (float)T_TOK;
    lt[e] = ALPHA_C * (float)NEXP * fi * Pi;
    __syncthreads();
    if (e == 0) {
        float s = 0.f;
        for (int i = 0; i < 64; ++i) s += lt[i];
        lossOut[0] = s;
    }
}

// =====================================================================
// Kernel C: ordered per-expert rank (token order!), builds slot->token
// inverse map for slots under capacity. Block = 128 tokens (4 waves).
// =====================================================================
__global__ __launch_bounds__(128) void moe_rank(
    const int* __restrict__ eIdx, const int* __restrict__ chunkBase,
    int* __restrict__ tok)
{
    __shared__ int waveHist[4][64];
    __shared__ int waveBase[4][64];
    const int tid = threadIdx.x, wave = tid >> 5, lane = tid & 31;
    const int t = blockIdx.x * 128 + tid;
    const int e = eIdx[t];

    int rank = 0, cnt = 0;
    for (int j = 0; j < 32; ++j) {
        int ej = __shfl(e, j);
        bool same = (ej == e);
        rank += (same && j < lane) ? 1 : 0;
        cnt  += same ? 1 : 0;
    }

    for (int i = tid; i < 256; i += 128) (&waveHist[0][0])[i] = 0;
    __syncthreads();
    if (rank == 0) waveHist[wave][e] = cnt;   // unique writer per (wave, expert)
    __syncthreads();
    if (tid < 64) {
        int bb = 0;
        #pragma unroll
        for (int w = 0; w < 4; ++w) { waveBase[w][tid] = bb; bb += waveHist[w][tid]; }
    }
    __syncthreads();

    int pos = chunkBase[blockIdx.x * 64 + e] + waveBase[wave][e] + rank;
    if (pos < CAP) tok[e * CAP + pos] = t;
}

// =====================================================================
// Kernel D: one block per output slot (64*2560 slots, 4KB each).
// Writes every slot exactly once: token copy or zeros. Coalesced b32
// (output is misaligned by 4B due to the leading loss scalar).
// =====================================================================
__global__ __launch_bounds__(256) void moe_gather(
    const float* __restrict__ x, const int* __restrict__ tok,
    const int* __restrict__ filled, float* __restrict__ outB)
{
    const int s = blockIdx.x;
    const int e = s / CAP;
    const int p = s - e * CAP;
    const int tid = threadIdx.x;
    float* o = outB + (size_t)s * DIM;
    if (p < filled[e]) {
        const float* src = x + (size_t)tok[s] * DIM;
        #pragma unroll
        for (int k = 0; k < 4; ++k) o[k * 256 + tid] = src[k * 256 + tid];
    } else {
        #pragma unroll
        for (int k = 0; k < 4; ++k) o[k * 256 + tid] = 0.f;
    }
}

extern "C" void kernel_launch(void* const* d_in, const int* in_sizes, int n_in,
                              void* d_out, int out_size, void* d_ws, size_t ws_size,
                              hipStream_t stream) {
    const float* x = (const float*)d_in[0];   // (131072, 1024)
    const float* W = (const float*)d_in[1];   // (1024, 64)
    const float* b = (const float*)d_in[2];   // (64,)
    float* out = (float*)d_out;               // [loss | out_batch(64,2560,1024)]

    char* ws = (char*)d_ws;
    int*   eIdx        = (int*)ws;   ws += (size_t)T_TOK * 4;
    int*   chunkCounts = (int*)ws;   ws += 1024 * 64 * 4;
    int*   chunkBase   = (int*)ws;   ws += 1024 * 64 * 4;
    float* piPart      = (float*)ws; ws += 1024 * 64 * 4;
    int*   filled      = (int*)ws;   ws += 64 * 4;
    int*   tok         = (int*)ws;   ws += (size_t)NEXP * CAP * 4;

    moe_router_gemm<<<1024, 256, 0, stream>>>(x, W, b, eIdx, chunkCounts, piPart);
    moe_scan<<<1, 64, 0, stream>>>(chunkCounts, piPart, chunkBase, filled, out);
    moe_rank<<<1024, 128, 0, stream>>>(eIdx, chunkBase, tok);
    moe_gather<<<NEXP * CAP, 256, 0, stream>>>(x, tok, filled, out + 1);
}